// MinLSTM_87222195848297
// MI455X (gfx1250) — compile-verified
//
#include <hip/hip_runtime.h>
#include <hip/hip_bf16.h>
#include <math.h>

// Problem dims (fixed by reference setup_inputs)
#define B_      8
#define S_      8192
#define D_      512
#define H_      256
#define CH_LEN  64
#define NCHUNK  (S_ / CH_LEN)          // 128
#define NCHANNEL (B_ * H_)             // 2048
#define NAGG    (B_ * H_ * NCHUNK)     // 262144

typedef __attribute__((ext_vector_type(2))) float v2f;
typedef __attribute__((ext_vector_type(8))) float v8f;

__device__ __forceinline__ float softplus_f(float x) {
    // stable: max(x,0) + log1p(exp(-|x|))
    return fmaxf(x, 0.0f) + log1pf(expf(-fabsf(x)));
}
__device__ __forceinline__ float log_g(float z) {
    // log(z + 0.5) if z >= 0 else log(sigmoid(z)) = -softplus(-z)
    return (z >= 0.0f) ? logf(z + 0.5f) : -softplus_f(-z);
}
__device__ __forceinline__ float logaddexp_f(float x, float y) {
    float m = fmaxf(x, y);
    if (m == -INFINITY) return -INFINITY;
    return m + log1pf(expf(-fabsf(x - y)));
}

// ---------------------------------------------------------------------------
// Kernel 1: fused GEMM (y = x @ W^T via V_WMMA_F32_16X16X4_F32) + gate math.
// One wave computes one (16-row M-tile, 16-col H-tile): three matched 16x16
// output tiles (f, i, h~ columns of W) sharing the A fragment.
// Writes log_f -> lf (ws), log_v = log_i + log_g(h~) -> lv (staged in d_out).
// ---------------------------------------------------------------------------
__global__ void __launch_bounds__(128) minlstm_gemm_gates(
        const float* __restrict__ x,   // [B*S, D]
        const float* __restrict__ W,   // [3H, D]
        float* __restrict__ lf,        // [B*S, H]
        float* __restrict__ lv) {      // [B*S, H]
    const int lane  = threadIdx.x & 31;
    const int wid   = blockIdx.x * (blockDim.x >> 5) + (threadIdx.x >> 5);
    const int mtile = wid >> 4;          // 0..4095
    const int htile = wid & 15;          // 0..15
    const int m0 = mtile * 16;
    const int h0 = htile * 16;
    const int lo = lane & 15;
    const int hi = lane >> 4;
    const int koff = 2 * hi;             // lanes 0-15 hold K={0,1}, 16-31 K={2,3}

    const float* xrow = x + (size_t)(m0 + lo) * D_;
    const float* wf   = W + (size_t)(h0 + lo) * D_;            // f gate rows
    const float* wi   = W + (size_t)(H_ + h0 + lo) * D_;       // i gate rows
    const float* wh   = W + (size_t)(2 * H_ + h0 + lo) * D_;   // h~ rows

    v8f accf = {}, acci = {}, acch = {};
#pragma unroll 4
    for (int k0 = 0; k0 < D_; k0 += 4) {
        v2f a  = *(const v2f*)(xrow + k0 + koff);
        v2f bf = *(const v2f*)(wf + k0 + koff);
        v2f bi = *(const v2f*)(wi + k0 + koff);
        v2f bh = *(const v2f*)(wh + k0 + koff);
        accf = __builtin_amdgcn_wmma_f32_16x16x4_f32(
            false, a, false, bf, (short)0, accf, false, false);
        acci = __builtin_amdgcn_wmma_f32_16x16x4_f32(
            false, a, false, bi, (short)0, acci, false, false);
        acch = __builtin_amdgcn_wmma_f32_16x16x4_f32(
            false, a, false, bh, (short)0, acch, false, false);
    }

    // C/D layout: VGPR r -> row m0 + r + 8*hi, col h0 + lo
#pragma unroll
    for (int r = 0; r < 8; ++r) {
        const int m = m0 + r + 8 * hi;
        const size_t idx = (size_t)m * H_ + (h0 + lo);
        const float f  = accf[r];
        const float i  = acci[r];
        const float hh = acch[r];
        const float diff   = softplus_f(-f) - softplus_f(-i);
        const float log_fg = -softplus_f(diff);
        const float log_ig = -softplus_f(-diff);
        lf[idx] = log_fg;
        lv[idx] = log_ig + log_g(hh);
    }
}

// ---------------------------------------------------------------------------
// Kernel 2: per-chunk aggregate of the associative scan operator.
// Element e_s = (lf_s, lv_s); state (A,B) ∘ e = (A+lf, logaddexp(B+lf, lv)).
// Each thread folds one chunk of 64 steps starting from identity (0, -inf).
// ---------------------------------------------------------------------------
__global__ void __launch_bounds__(256) minlstm_chunk_agg(
        const float* __restrict__ lf,
        const float* __restrict__ lv,
        float* __restrict__ agg_a,
        float* __restrict__ agg_b) {
    const int t = blockIdx.x * blockDim.x + threadIdx.x;  // 262144
    const int h = t & (H_ - 1);
    const int c = (t >> 8) & (NCHUNK - 1);
    const int b = t >> 15;
    const size_t base = ((size_t)b * S_ + (size_t)c * CH_LEN) * H_ + h;
    float A = 0.0f, Bv = -INFINITY;
    for (int s = 0; s < CH_LEN; ++s) {
        const float f = lf[base + (size_t)s * H_];
        const float v = lv[base + (size_t)s * H_];
        A += f;
        Bv = logaddexp_f(Bv + f, v);
    }
    const size_t ai = ((size_t)b * NCHUNK + c) * H_ + h;
    agg_a[ai] = A;
    agg_b[ai] = Bv;
}

// ---------------------------------------------------------------------------
// Kernel 3: serial scan over the 128 chunk aggregates per channel, seeded
// with log_g(h_prev). Stores the incoming log_h carry for each chunk.
// ---------------------------------------------------------------------------
__global__ void __launch_bounds__(256) minlstm_chunk_scan(
        const float* __restrict__ h_prev,   // [B, H]
        const float* __restrict__ agg_a,
        const float* __restrict__ agg_b,
        float* __restrict__ carry) {
    const int t = blockIdx.x * blockDim.x + threadIdx.x;  // 2048
    const int h = t & (H_ - 1);
    const int b = t >> 8;
    float Bv = log_g(h_prev[(size_t)b * H_ + h]);
    for (int c = 0; c < NCHUNK; ++c) {
        const size_t ai = ((size_t)b * NCHUNK + c) * H_ + h;
        carry[ai] = Bv;
        Bv = logaddexp_f(Bv + agg_a[ai], agg_b[ai]);
    }
}

// ---------------------------------------------------------------------------
// Kernel 4: re-scan each chunk with its carry; emit h_t = exp(log_h_t).
// Reads lv from d_out and overwrites the same slot (read-before-write per
// element within the owning thread), so d_out doubles as the lv stage.
// ---------------------------------------------------------------------------
__global__ void __launch_bounds__(256) minlstm_chunk_apply(
        const float* __restrict__ lf,
        const float* __restrict__ carry,
        float* __restrict__ lv_out) {
    const int t = blockIdx.x * blockDim.x + threadIdx.x;  // 262144
    const int h = t & (H_ - 1);
    const int c = (t >> 8) & (NCHUNK - 1);
    const int b = t >> 15;
    const size_t base = ((size_t)b * S_ + (size_t)c * CH_LEN) * H_ + h;
    float Bv = carry[((size_t)b * NCHUNK + c) * H_ + h];
    for (int s = 0; s < CH_LEN; ++s) {
        const size_t idx = base + (size_t)s * H_;
        Bv = logaddexp_f(Bv + lf[idx], lv_out[idx]);
        lv_out[idx] = expf(Bv);
    }
}

extern "C" void kernel_launch(void* const* d_in, const int* in_sizes, int n_in,
                              void* d_out, int out_size, void* d_ws, size_t ws_size,
                              hipStream_t stream) {
    const float* x      = (const float*)d_in[0];  // [B,S,D]
    const float* h_prev = (const float*)d_in[1];  // [B,H]
    const float* W      = (const float*)d_in[2];  // [3H,D]
    float* out = (float*)d_out;                   // [B,S,H]

    // Workspace layout (floats): lf (64MB) | agg_a | agg_b | carry (1MB each)
    float* lf    = (float*)d_ws;
    float* agg_a = lf + (size_t)B_ * S_ * H_;
    float* agg_b = agg_a + NAGG;
    float* carry = agg_b + NAGG;

    // 1) WMMA GEMM + gates: (65536/16)*(256/16) = 65536 waves, 4 waves/block
    const int gemm_blocks = (B_ * S_ / 16) * (H_ / 16) / 4;  // 16384
    minlstm_gemm_gates<<<gemm_blocks, 128, 0, stream>>>(x, W, lf, out);

    // 2) chunk aggregates: 262144 threads
    minlstm_chunk_agg<<<NAGG / 256, 256, 0, stream>>>(lf, out, agg_a, agg_b);

    // 3) carry scan across chunks: 2048 threads
    minlstm_chunk_scan<<<NCHANNEL / 256, 256, 0, stream>>>(h_prev, agg_a, agg_b, carry);

    // 4) final per-chunk apply, writes exp(log_h) into d_out
    minlstm_chunk_apply<<<NAGG / 256, 256, 0, stream>>>(lf, carry, out);
}